// SampledFBP_82317343195728
// MI455X (gfx1250) — compile-verified
//
#include <hip/hip_runtime.h>

namespace {
constexpr int kN  = 2;     // batch
constexpr int kA  = 360;   // angles
constexpr int kM  = 512;   // detector bins
constexpr int kNP = 32;    // patches
constexpr int kPS = 64;    // patch size
constexpr int kQSplit  = 4;                      // row splits per patch -> 128 blocks
constexpr int kThreads = 256;                    // 8 wave32 per block
constexpr int kRows    = kPS / kQSplit;          // 16 rows per block
constexpr int kPix     = kRows * kPS / kThreads; // 4 pixels per thread
constexpr int kChunk   = 6;                      // angles staged per chunk
constexpr int kNChunks = kA / kChunk;            // 60
// LDS staging: [buf(2)][n(2)][aa(kChunk)][m(kM)] floats
constexpr int kBufF    = kN * kChunk * kM;       // 6144 floats per buffer
constexpr int kStageF  = 2 * kBufF;              // 12288 floats = 48 KB
constexpr unsigned kBufBytes   = (unsigned)(kBufF * 4);      // 24576
constexpr unsigned kAngleBytes = (unsigned)(kM * 4);         // 2048
constexpr unsigned kNPlaneG    = (unsigned)(kA * kM * 4);    // global n stride
}

template<int IMM>
__device__ __forceinline__ void async_stage(unsigned ldsdst, unsigned voff,
                                            const float* saddr) {
  // offset: is added to BOTH the LDS destination and the global source
  // (CDNA5 ISA) -> LDS layout mirrors the 2048B angle stride.
  asm volatile("global_load_async_to_lds_b128 %0, %1, %2 offset:%3"
               :: "v"(ldsdst), "v"(voff), "s"(saddr), "n"(IMM) : "memory");
}

__device__ __forceinline__ void issue_chunk(unsigned ldsdst, unsigned voff,
                                            const float* saddr) {
  async_stage<0>(ldsdst, voff, saddr);
  async_stage<2048>(ldsdst, voff, saddr);
  async_stage<4096>(ldsdst, voff, saddr);
  async_stage<6144>(ldsdst, voff, saddr);
  async_stage<8192>(ldsdst, voff, saddr);
  async_stage<10240>(ldsdst, voff, saddr);
}

__global__ __launch_bounds__(kThreads)
void fbp_backproject(const float* __restrict__ sino,
                     const float* __restrict__ theta,
                     const float* __restrict__ cx,
                     const float* __restrict__ cy,
                     float* __restrict__ out)
{
  __shared__ float2 trig[kA];        // {cos, sin} per angle, 2.9 KB
  __shared__ float  stage[kStageF];  // 48 KB double-buffered staging

  const int t = (int)threadIdx.x;
  const int p = (int)blockIdx.x >> 2;   // patch id
  const int q = (int)blockIdx.x & 3;    // row-split id

  // One-time trig table: all per-angle transcendentals leave the hot loop.
  for (int idx = t; idx < kA; idx += kThreads) {
    float s, c;
    sincosf(theta[idx], &s, &c);
    trig[idx] = make_float2(c, s);
  }

  const float half = 0.5f * (float)(kM - 1);  // 255.5
  const float cxp  = cx[p];
  const float cyp  = cy[p];

  const int j  = t & (kPS - 1);                 // column 0..63
  const int ib = q * kRows + (t >> 6) * kPix;   // first of 4 contiguous rows

  // u(i,j,a) = ct*Xi - st*Yj + half  (algebraically identical to reference)
  const float Yj = (float)(j - kPS / 2) + cyp - half;

  float Xi[kPix];
  float msk[kPix];
  #pragma unroll
  for (int k = 0; k < kPix; ++k) {
    Xi[k]  = (float)(ib + k - kPS / 2) + cxp - half;
    msk[k] = (Xi[k] * Xi[k] + Yj * Yj <= half * half) ? 1.0f : 0.0f;
  }

  float acc0[kPix] = {0.0f, 0.0f, 0.0f, 0.0f};  // n = 0
  float acc1[kPix] = {0.0f, 0.0f, 0.0f, 0.0f};  // n = 1

  // ---- async staging addresses ------------------------------------------
  // Thread t moves 16B of each angle-row: n = t>>7 selects the batch plane,
  // (t&127)*16 is the byte position within a 2 KB row.
  const int nPart = t >> 7;
  const int mOff  = (t & 127) * 16;
  const unsigned stagebase = (unsigned)(unsigned long long)(const void*)stage;
  const unsigned dstBase   = stagebase
                           + (unsigned)(nPart * (kChunk * (int)kAngleBytes))
                           + (unsigned)mOff;
  unsigned voff = (nPart ? kNPlaneG : 0u) + (unsigned)mOff;

  issue_chunk(dstBase, voff, sino);          // chunk 0 -> buffer 0
  voff += kChunk * kAngleBytes;

  for (int c = 0; c < kNChunks; ++c) {
    const int buf = c & 1;
    if (c + 1 < kNChunks) {
      issue_chunk(dstBase + (unsigned)((buf ^ 1)) * kBufBytes, voff, sino);
      voff += kChunk * kAngleBytes;
      // 6 newly outstanding; <=6 left  =>  previous chunk fully in LDS
      asm volatile("s_wait_asynccnt 6" ::: "memory");
    } else {
      asm volatile("s_wait_asynccnt 0" ::: "memory");
    }
    __syncthreads();  // all waves' slices (and, at c==0, trig table) visible

    const int fb = buf * kBufF;  // float base of current buffer
    #pragma unroll 2
    for (int aa = 0; aa < kChunk; ++aa) {
      const float2 cs = trig[c * kChunk + aa];  // broadcast ds_load_b64
      const float  c0 = half - cs.y * Yj;
      const int    r0 = fb + aa * kM;           // n=0 row (floats)
      #pragma unroll
      for (int k = 0; k < kPix; ++k) {
        const float u  = fmaf(cs.x, Xi[k], c0);
        const float fi = floorf(u);
        const float w  = u - fi;
        const int   i0 = (int)fi;
        const int   i1 = i0 + 1;
        const float w0 = (i0 >= 0 && i0 < kM) ? (1.0f - w) : 0.0f;
        const float w1 = (i1 >= 0 && i1 < kM) ? w : 0.0f;
        const int i0c = i0 < 0 ? 0 : (i0 > kM - 1 ? kM - 1 : i0);
        const int i1c = i1 < 0 ? 0 : (i1 > kM - 1 ? kM - 1 : i1);
        const float s00 = stage[r0 + i0c];
        const float s01 = stage[r0 + i1c];
        const float s10 = stage[r0 + kChunk * kM + i0c];  // n=1 plane
        const float s11 = stage[r0 + kChunk * kM + i1c];
        acc0[k] = fmaf(w0, s00, fmaf(w1, s01, acc0[k]));
        acc1[k] = fmaf(w0, s10, fmaf(w1, s11, acc1[k]));
      }
    }
    __syncthreads();  // done reading this buffer; safe to overwrite
  }

  const float scale = 3.14159265358979323846f / (2.0f * (float)kA);
  #pragma unroll
  for (int k = 0; k < kPix; ++k) {
    const int   i = ib + k;
    const float m = msk[k] * scale;
    out[((0 * kNP + p) * kPS + i) * kPS + j] = acc0[k] * m;  // coalesced rows
    out[((1 * kNP + p) * kPS + i) * kPS + j] = acc1[k] * m;
  }
}

extern "C" void kernel_launch(void* const* d_in, const int* in_sizes, int n_in,
                              void* d_out, int out_size, void* d_ws, size_t ws_size,
                              hipStream_t stream) {
  (void)in_sizes; (void)n_in; (void)d_ws; (void)ws_size; (void)out_size;
  const float* sino  = (const float*)d_in[0];  // (2,1,360,512) f32, pre-filtered
  const float* theta = (const float*)d_in[1];  // (360,)
  const float* cxs   = (const float*)d_in[2];  // (32,)
  const float* cys   = (const float*)d_in[3];  // (32,)
  float* out = (float*)d_out;                  // (2,32,64,64) f32

  dim3 grid(kNP * kQSplit);   // 128 blocks
  dim3 block(kThreads);       // 256 threads = 8 wave32
  hipLaunchKernelGGL(fbp_backproject, grid, block, 0, stream,
                     sino, theta, cxs, cys, out);
}